// RegionProposal_5291399708892
// MI455X (gfx1250) — compile-verified
//
#include <hip/hip_runtime.h>
#include <stdint.h>

// ---------------------------------------------------------------------------
// RegionProposal post-processing for MI455X (gfx1250, wave32).
// Latency/launch-bound problem (~2.2MB inputs, fits in L2 many times over):
// fuse passes, stage everything in LDS, use the gfx1250 async global->LDS
// copy engine (global_load_async_to_lds_b128 + s_wait_asynccnt) and
// global_prefetch_b8. No WMMA: the workload has no bilinear structure.
// ---------------------------------------------------------------------------

#define A_      9
#define H_      64
#define W_      96
#define N_      55296      // H*W*A anchors
#define NPAD_   65536      // next pow2 for bitonic sort
#define TOPK_   6000       // PRE_NMS_TOP_K
#define OUTK_   300        // POST_NMS_TOP_K
#define IOU_TH  0.7f
#define NWORDS_ 94         // ceil(TOPK_/64) 64-bit mask words per row

typedef unsigned long long u64;
typedef unsigned int       u32;

#if defined(__has_builtin)
# if __has_builtin(__builtin_amdgcn_global_load_async_to_lds_b128) && \
     __has_builtin(__builtin_amdgcn_s_wait_asynccnt)
#  define USE_ASYNC_LDS 1
# endif
#endif
#ifndef USE_ASYNC_LDS
# define USE_ASYNC_LDS 0
#endif

#if USE_ASYNC_LDS
typedef int v4i __attribute__((vector_size(16)));
typedef __attribute__((address_space(1))) v4i* as1_v4i;
typedef __attribute__((address_space(3))) v4i* as3_v4i;

// Async copy of one 16B chunk global -> LDS. Flat->AS pointers go through
// integers: AS1 numeric value == 64-bit VA; AS3 numeric value == low 32 bits
// of the flat address, which per the gfx1250 aperture layout IS the LDS offset.
__device__ __forceinline__ void async_copy_b128(const void* gsrc, void* ldst) {
  __builtin_amdgcn_global_load_async_to_lds_b128(
      (as1_v4i)(unsigned long long)gsrc,
      (as3_v4i)(u32)(unsigned long long)ldst,
      0, 0);
}
__device__ __forceinline__ void async_wait0() {
  __builtin_amdgcn_s_wait_asynccnt(0);
}
#endif

// Monotonic float->uint mapping (ascending order preserved).
__device__ __forceinline__ u32 fkey(float f) {
  u32 u = __float_as_uint(f);
  return (u & 0x80000000u) ? ~u : (u | 0x80000000u);
}

// ---------------------------------------------------------------------------
// Kernel 1: NCHW gather + sigmoid score + box decode + clip + sort-key pack.
// ---------------------------------------------------------------------------
__global__ void rpn_decode(const float* __restrict__ cls,
                           const float* __restrict__ loc,
                           const float* __restrict__ anc,
                           u64* __restrict__ keys,
                           float4* __restrict__ boxes) {
  int n = blockIdx.x * blockDim.x + threadIdx.x;
  if (n >= NPAD_) return;
  if (n >= N_) { keys[n] = 0ull; return; }   // padding sorts to the end (desc)

  const int HW = H_ * W_;
  int a  = n % A_;
  int hw = n / A_;
  int w  = hw % W_;
  int h  = hw / W_;
  int sp = h * W_ + w;

  // cls channel (a*2 + 1) -> foreground logit; sigmoid.
  float lg    = cls[(a * 2 + 1) * HW + sp];
  float score = 1.f / (1.f + expf(-lg));

  float t0 = loc[(a * 4 + 0) * HW + sp];
  float t1 = loc[(a * 4 + 1) * HW + sp];
  float t2 = loc[(a * 4 + 2) * HW + sp];
  float t3 = loc[(a * 4 + 3) * HW + sp];

  float acx = anc[n * 4 + 0], acy = anc[n * 4 + 1];
  float aw  = anc[n * 4 + 2], ah  = anc[n * 4 + 3];

  float cx = t0 * aw + acx;
  float cy = t1 * ah + acy;
  float bw = expf(t2) * aw;
  float bh = expf(t3) * ah;

  float x1 = fminf(fmaxf(cx - 0.5f * bw, 0.f), 1.f);
  float y1 = fminf(fmaxf(cy - 0.5f * bh, 0.f), 1.f);
  float x2 = fminf(fmaxf(cx + 0.5f * bw, 0.f), 1.f);
  float y2 = fminf(fmaxf(cy + 0.5f * bh, 0.f), 1.f);

  boxes[n] = make_float4(x1, y1, x2, y2);
  // High 32: monotonic score key; low 32: ~n so descending sort is "stable"
  // (ties -> smaller original index first), matching jnp.argsort(-scores).
  keys[n] = ((u64)fkey(score) << 32) | (u32)(~(u32)n);
}

// ---------------------------------------------------------------------------
// Shared helper: load a 2048-key tile into LDS (async path when available).
// ---------------------------------------------------------------------------
__device__ __forceinline__ void load_tile_2048(u64* __restrict__ s,
                                               const u64* __restrict__ keys,
                                               u32 base, u32 t) {
#if USE_ASYNC_LDS
  // Each of the 1024 threads pulls one 16B chunk (two keys) straight to LDS.
  async_copy_b128((const void*)(keys + base + 2u * t), (void*)(s + 2u * t));
  async_wait0();
  __syncthreads();
#else
  s[t]        = keys[base + t];
  s[t + 1024] = keys[base + t + 1024];
  __syncthreads();
#endif
}

// ---------------------------------------------------------------------------
// Kernel 2a: one global bitonic compare-exchange step (stride j > 1024).
// Descending sort overall.
// ---------------------------------------------------------------------------
__global__ void rpn_bitonic_step(u64* __restrict__ keys, u32 k, u32 j) {
  u32 i = blockIdx.x * blockDim.x + threadIdx.x;
  if (i >= NPAD_) return;
  u32 ixj = i ^ j;
  if (ixj <= i) return;
  u64 a = keys[i], b = keys[ixj];
  bool descBlock = ((i & k) == 0);
  bool sw = descBlock ? (a < b) : (a > b);
  if (sw) { keys[i] = b; keys[ixj] = a; }
}

// ---------------------------------------------------------------------------
// Kernel 2b: ENTIRE first sort phase (k=2..2048, every stride) in one launch,
// fully LDS-resident per 2048-element tile (16KB of the 320KB/WGP budget).
// Leaves tiles alternately desc/asc — the bitonic precondition for k=4096.
// ---------------------------------------------------------------------------
__global__ void rpn_bitonic_local(u64* __restrict__ keys) {
  __shared__ __align__(16) u64 s[2048];
  u32 base = blockIdx.x * 2048;
  u32 t = threadIdx.x;
  load_tile_2048(s, keys, base, t);
  for (u32 k = 2; k <= 2048; k <<= 1) {
    for (u32 j = k >> 1; j >= 1; j >>= 1) {
      u32 i  = ((t & ~(j - 1)) << 1) | (t & (j - 1));
      u32 gi = base + i;
      u64 a = s[i], b = s[i + j];
      bool descBlock = ((gi & k) == 0);
      bool sw = descBlock ? (a < b) : (a > b);
      if (sw) { s[i] = b; s[i + j] = a; }
      __syncthreads();
    }
  }
  keys[base + t]        = s[t];
  keys[base + t + 1024] = s[t + 1024];
}

// ---------------------------------------------------------------------------
// Kernel 2c: fused bitonic tail for one merge stage k: strides 1024..1 inside
// a 2048-element LDS tile.
// ---------------------------------------------------------------------------
__global__ void rpn_bitonic_fused(u64* __restrict__ keys, u32 k) {
  __shared__ __align__(16) u64 s[2048];
  u32 base = blockIdx.x * 2048;
  u32 t = threadIdx.x;
  load_tile_2048(s, keys, base, t);
  for (u32 j = 1024; j >= 1; j >>= 1) {
    u32 i  = ((t & ~(j - 1)) << 1) | (t & (j - 1));
    u32 gi = base + i;
    u64 a = s[i], b = s[i + j];
    bool descBlock = ((gi & k) == 0);
    bool sw = descBlock ? (a < b) : (a > b);
    if (sw) { s[i] = b; s[i + j] = a; }
    __syncthreads();
  }
  keys[base + t]        = s[t];
  keys[base + t + 1024] = s[t + 1024];
}

// ---------------------------------------------------------------------------
// Kernel 3: gather top-6000 boxes in descending-score order.
// ---------------------------------------------------------------------------
__global__ void rpn_gather(const u64* __restrict__ keys,
                           const float4* __restrict__ boxes,
                           float4* __restrict__ roi) {
  int t = blockIdx.x * blockDim.x + threadIdx.x;
  if (t >= TOPK_) return;
  u32 n = ~(u32)(keys[t] & 0xFFFFFFFFull);
  if (n >= N_) n = 0;   // cannot happen (N > TOPK), defensive
  roi[t] = boxes[n];
}

// ---------------------------------------------------------------------------
// Kernel 4: pairwise IoU suppression bitmask. Block = 64 rows x 64 cols.
// Column tile staged in LDS via gfx1250 async global->LDS copy.
// ---------------------------------------------------------------------------
__global__ void rpn_nms_mask(const float4* __restrict__ roi,
                             u64* __restrict__ mask) {
  __shared__ __align__(16) float4 shbox[64];
  int bx = blockIdx.x, by = blockIdx.y;
  int t   = threadIdx.x;
  int row = by * 64 + t;

  if (bx < by) {  // all cols < all rows in this tile -> no suppression bits
    if (row < TOPK_) mask[(size_t)row * NWORDS_ + bx] = 0ull;
    return;
  }

  int col0  = bx * 64;
  int ncols = TOPK_ - col0; if (ncols > 64) ncols = 64;

#if USE_ASYNC_LDS
  if (t < ncols)
    async_copy_b128((const void*)(roi + col0 + t), (void*)(&shbox[t]));
  async_wait0();
  __syncthreads();
#else
  if (t < ncols) shbox[t] = roi[col0 + t];
  __syncthreads();
#endif

  if (row >= TOPK_) return;
  float4 r  = roi[row];
  float  ra = (r.z - r.x) * (r.w - r.y);

  u64 bits = 0ull;
  for (int b = 0; b < ncols; ++b) {
    int col = col0 + b;
    if (col <= row) continue;
    float4 c  = shbox[b];
    float xx1 = fmaxf(r.x, c.x), yy1 = fmaxf(r.y, c.y);
    float xx2 = fminf(r.z, c.z), yy2 = fminf(r.w, c.w);
    float iw  = fmaxf(xx2 - xx1, 0.f);
    float ih  = fmaxf(yy2 - yy1, 0.f);
    float inter = iw * ih;
    float ca  = (c.z - c.x) * (c.w - c.y);
    float uni = ra + ca - inter;
    if (inter > IOU_TH * uni) bits |= (1ull << b);
  }
  mask[(size_t)row * NWORDS_ + bx] = bits;
}

// ---------------------------------------------------------------------------
// Kernel 5: serial greedy reduce over 6000 rows by ONE wave32 workgroup
// (single-wave barriers are S_NOP per ISA). Suppression bitset lives in LDS.
// Emits first 300 kept boxes in order, zero-pads the remainder.
// ---------------------------------------------------------------------------
__global__ void rpn_nms_reduce(const u64* __restrict__ mask,
                               const float4* __restrict__ roi,
                               float* __restrict__ out) {
  __shared__ u64 remv[96];
  int lane = threadIdx.x;  // 0..31
  for (int t = lane; t < 96; t += 32) remv[t] = 0ull;
  __syncthreads();

  int outcount = 0;   // uniform across the wave
  for (int i = 0; i < TOPK_; ++i) {
    if (i + 1 < TOPK_)
      __builtin_prefetch(mask + (size_t)(i + 1) * NWORDS_, 0, 0);  // global_prefetch_b8
    u64 rm = remv[i >> 6];
    bool keep = ((rm >> (i & 63)) & 1ull) == 0ull;
    if (keep) {
      for (int t = lane; t < NWORDS_; t += 32)
        remv[t] |= mask[(size_t)i * NWORDS_ + t];
      if (outcount < OUTK_ && lane == 0)
        ((float4*)out)[outcount] = roi[i];
      ++outcount;
    }
    __syncthreads();
  }

  int oc = outcount < OUTK_ ? outcount : OUTK_;
  for (int m = oc * 4 + lane; m < OUTK_ * 4; m += 32) out[m] = 0.f;
}

// ---------------------------------------------------------------------------
// Host-side orchestration (all on `stream`, graph-capture safe, 25 launches).
// ---------------------------------------------------------------------------
extern "C" void kernel_launch(void* const* d_in, const int* in_sizes, int n_in,
                              void* d_out, int out_size, void* d_ws, size_t ws_size,
                              hipStream_t stream) {
  (void)in_sizes; (void)n_in; (void)out_size; (void)ws_size;
  const float* pred_cls = (const float*)d_in[0];
  const float* pred_loc = (const float*)d_in[1];
  const float* anchor   = (const float*)d_in[2];
  float* out = (float*)d_out;

  // Workspace layout (all 16B aligned):
  char* ws = (char*)d_ws;
  u64*    keys  = (u64*)(ws);                                   // 65536 * 8  = 512KB
  float4* boxes = (float4*)(ws + (size_t)NPAD_ * 8);            // 55296 * 16 = 864KB
  float4* roi   = (float4*)(ws + (size_t)NPAD_ * 8
                               + (size_t)N_ * 16);              // 6000 * 16  = 94KB
  u64*    mask  = (u64*)(ws + (size_t)NPAD_ * 8
                            + (size_t)N_ * 16
                            + (size_t)TOPK_ * 16);              // 6000*94*8 ~= 4.3MB

  // 1) decode + score + key pack
  rpn_decode<<<NPAD_ / 256, 256, 0, stream>>>(pred_cls, pred_loc, anchor, keys, boxes);

  // 2) full descending bitonic sort of 65536 keys.
  //    Phase A: k=2..2048 entirely in LDS (one launch).
  rpn_bitonic_local<<<NPAD_ / 2048, 1024, 0, stream>>>(keys);
  //    Phase B: k=4096..65536 — global steps for strides >1024, LDS tail.
  for (u32 k = 4096; k <= (u32)NPAD_; k <<= 1) {
    for (u32 j = k >> 1; j > 1024; j >>= 1)
      rpn_bitonic_step<<<NPAD_ / 256, 256, 0, stream>>>(keys, k, j);
    rpn_bitonic_fused<<<NPAD_ / 2048, 1024, 0, stream>>>(keys, k);
  }

  // 3) gather top-6000 boxes
  rpn_gather<<<(TOPK_ + 255) / 256, 256, 0, stream>>>(keys, boxes, roi);

  // 4) pairwise IoU bitmask
  dim3 g(NWORDS_, NWORDS_);
  rpn_nms_mask<<<g, 64, 0, stream>>>(roi, mask);

  // 5) greedy reduce + emit [300,4]
  rpn_nms_reduce<<<1, 32, 0, stream>>>(mask, roi, out);
}